// NaiveMHCLayer_44023414784001
// MI455X (gfx1250) — compile-verified
//
#include <hip/hip_runtime.h>
#include <hip/hip_bf16.h>
#include <stdint.h>

// ---------------------------------------------------------------------------
// NaiveMHCLayer for MI455X (gfx1250), B=4096, n=4, C=4096, fp32.
// Bandwidth-bound (~512 MiB traffic floor): read x once via async
// global->LDS, one WMMA 16x16x4_f32 per 16-column tile computes
//   D = M_pad(16x4) @ x(4x16) + y_dist ,
// tile-pairs are shuffle-combined so every store is a full-wave 128B burst.
// ---------------------------------------------------------------------------

#define N_EXP 4
#define C_DIM 4096
#define BLOCK_THREADS 256
#define MHC_EPS 1e-5f

typedef __attribute__((ext_vector_type(2))) float v2f;
typedef __attribute__((ext_vector_type(4))) float v4f;
typedef __attribute__((ext_vector_type(8))) float v8f;
typedef __attribute__((ext_vector_type(4))) int   v4i;

#define GLOBAL_AS __attribute__((address_space(1)))
#define LOCAL_AS  __attribute__((address_space(3)))

#if defined(__has_builtin)
#if __has_builtin(__builtin_amdgcn_global_load_async_to_lds_b128)
#define HAVE_ASYNC_LDS 1
#endif
#if __has_builtin(__builtin_amdgcn_s_wait_asynccnt)
#define HAVE_WAIT_ASYNC 1
#endif
#endif

__device__ __forceinline__ void wait_asynccnt0() {
#if defined(HAVE_WAIT_ASYNC)
  __builtin_amdgcn_s_wait_asynccnt(0);
#elif defined(HAVE_ASYNC_LDS)
  asm volatile("s_wait_asynccnt 0" ::: "memory");
#endif
}

// ---------------------------------------------------------------------------
// Kernel 1: gates + Sinkhorn-Knopp on the 4x4 matrix. Trivial scalar work.
// ws layout (floats): [0..3]=sigmoid(H_pre), [4..7]=2*sigmoid(H_post),
//                     [8..23]=M row-major.
// ---------------------------------------------------------------------------
__global__ void mhc_prep_kernel(const float* __restrict__ Hpre,
                                const float* __restrict__ Hpost,
                                const float* __restrict__ Hres,
                                float* __restrict__ ws) {
  if (threadIdx.x != 0 || blockIdx.x != 0) return;
  float M[4][4];
#pragma unroll
  for (int i = 0; i < 4; ++i)
#pragma unroll
    for (int j = 0; j < 4; ++j)
      M[i][j] = __expf(Hres[i * 4 + j]);

#pragma unroll 1
  for (int it = 0; it < 20; ++it) {
#pragma unroll
    for (int i = 0; i < 4; ++i) {
      float inv = 1.0f / (M[i][0] + M[i][1] + M[i][2] + M[i][3] + MHC_EPS);
      M[i][0] *= inv; M[i][1] *= inv; M[i][2] *= inv; M[i][3] *= inv;
    }
#pragma unroll
    for (int j = 0; j < 4; ++j) {
      float inv = 1.0f / (M[0][j] + M[1][j] + M[2][j] + M[3][j] + MHC_EPS);
      M[0][j] *= inv; M[1][j] *= inv; M[2][j] *= inv; M[3][j] *= inv;
    }
  }
#pragma unroll
  for (int i = 0; i < 4; ++i) {
    ws[i]     = 1.0f / (1.0f + __expf(-Hpre[i]));
    ws[4 + i] = 2.0f / (1.0f + __expf(-Hpost[i]));
  }
#pragma unroll
  for (int i = 0; i < 4; ++i)
#pragma unroll
    for (int j = 0; j < 4; ++j)
      ws[8 + i * 4 + j] = M[i][j];
}

// ---------------------------------------------------------------------------
// Kernel 2: one block per batch row b.
// ---------------------------------------------------------------------------
__launch_bounds__(BLOCK_THREADS)
__global__ void mhc_main_kernel(const float* __restrict__ x,
                                const float* __restrict__ wgt,
                                const float* __restrict__ ws,
                                float* __restrict__ out) {
  __shared__ float xs[N_EXP * C_DIM];   // 64 KB stage of x[b]
  __shared__ float red[12];

  const int tid = threadIdx.x;
  const int b = blockIdx.x;
  const size_t xbase = (size_t)b * (N_EXP * C_DIM);

  // ---- Phase 1: stage x[b,:,:] into LDS (async global->LDS on gfx1250) ----
  {
    const float* gsrc = x + xbase;
#if defined(HAVE_ASYNC_LDS)
#pragma unroll 4
    for (int i4 = tid; i4 < (N_EXP * C_DIM) / 4; i4 += BLOCK_THREADS) {
      __builtin_amdgcn_global_load_async_to_lds_b128(
          (GLOBAL_AS v4i*)(gsrc + i4 * 4),
          (LOCAL_AS v4i*)(&xs[i4 * 4]),
          0, 0);
    }
    wait_asynccnt0();
#else
#pragma unroll 4
    for (int i4 = tid; i4 < (N_EXP * C_DIM) / 4; i4 += BLOCK_THREADS) {
      *(v4f*)(&xs[i4 * 4]) = *(const v4f*)(gsrc + i4 * 4);
    }
#endif
  }
  __syncthreads();

  const float pre0 = ws[0], pre1 = ws[1], pre2 = ws[2], pre3 = ws[3];

  // ---- Phase 2: sum over C of agg^2 for RMSNorm ----
  float ss = 0.0f;
#pragma unroll 4
  for (int c = tid; c < C_DIM; c += BLOCK_THREADS) {
    float agg = pre0 * xs[c]
              + pre1 * xs[C_DIM + c]
              + pre2 * xs[2 * C_DIM + c]
              + pre3 * xs[3 * C_DIM + c];
    ss += agg * agg;
  }
#pragma unroll
  for (int off = 16; off > 0; off >>= 1)
    ss += __shfl_xor(ss, off, 32);
  const int lane = tid & 31;
  const int wv = tid >> 5;
  if (lane == 0) red[wv] = ss;
  __syncthreads();
  if (tid == 0) {
    float t = 0.0f;
#pragma unroll
    for (int i = 0; i < BLOCK_THREADS / 32; ++i) t += red[i];
    red[8] = rsqrtf(t * (1.0f / (float)C_DIM) + MHC_EPS);
  }
  __syncthreads();
  const float inv_rms = red[8];

  const float post0 = ws[4], post1 = ws[5], post2 = ws[6], post3 = ws[7];

  // ---- Phase 3: per-wave WMMA over pairs of 16-column tiles ----
  // A (16x4 f32, 2 VGPRs): lanes 0-15 row m hold K={0,1}; lanes 16-31 K={2,3}.
  const int col16 = lane & 15;
  const bool hi = (lane >= 16);
  float a0 = 0.0f, a1 = 0.0f;
  if (col16 < 4) {
    const int kb = hi ? 2 : 0;
    a0 = ws[8 + col16 * 4 + kb];
    a1 = ws[8 + col16 * 4 + kb + 1];
  }
  const v2f Av = {a0, a1};

  // One 16x16x4 WMMA for the 16 columns starting at (ctile + col16's tile).
  auto tile_wmma = [&](int c) -> v8f {
    const float x0 = xs[c];
    const float x1 = xs[C_DIM + c];
    const float x2 = xs[2 * C_DIM + c];
    const float x3 = xs[3 * C_DIM + c];

    // B (4x16 f32): lanes 0-15 hold rows K={0,1}; lanes 16-31 rows K={2,3}.
    v2f Bv;
    Bv[0] = hi ? x2 : x0;
    Bv[1] = hi ? x3 : x1;

    // y_dist as accumulator C: rows 0..3 live in VGPRs 0..3, lanes 0-15.
    const float agg = pre0 * x0 + pre1 * x1 + pre2 * x2 + pre3 * x3;
    const float normed = agg * inv_rms * wgt[c];
    const float gate = hi ? 0.0f : normed;
    v8f Cv = {};
    Cv[0] = post0 * gate;
    Cv[1] = post1 * gate;
    Cv[2] = post2 * gate;
    Cv[3] = post3 * gate;

    // D = M_pad @ x_tile + y_dist  (reference output lives in rows 0..3)
    return __builtin_amdgcn_wmma_f32_16x16x4_f32(
        false, Av, false, Bv, (short)0, Cv, false, false);
  };

  const int pairsPerWave = (C_DIM / 32) / (BLOCK_THREADS / 32); // 16
#pragma unroll 2
  for (int p = 0; p < pairsPerWave; ++p) {
    const int c0 = (wv * pairsPerWave + p) * 32;   // 32 columns per pair

    const v8f De = tile_wmma(c0 + col16);          // columns c0   .. c0+15
    const v8f Do = tile_wmma(c0 + 16 + col16);     // columns c0+16.. c0+31

    // Merge: lane L<16 -> De row r col c0+L ; lane L>=16 -> Do row r col c0+L.
    // Do rows live in lanes 0-15; shuffle them up by 16 so every lane stores,
    // giving four full-wave 128B contiguous bursts per pair.
#pragma unroll
    for (int r = 0; r < 4; ++r) {
      const float oddUp = __shfl_up(Do[r], 16, 32);
      const float val = hi ? oddUp : De[r];
      out[xbase + (size_t)r * C_DIM + c0 + lane] = val;
    }
  }
}

// ---------------------------------------------------------------------------
extern "C" void kernel_launch(void* const* d_in, const int* in_sizes, int n_in,
                              void* d_out, int out_size, void* d_ws, size_t ws_size,
                              hipStream_t stream) {
  const float* x     = (const float*)d_in[0];  // (B, 4, 4096)
  const float* wgt   = (const float*)d_in[1];  // (4096,)
  const float* Hpre  = (const float*)d_in[2];  // (4,)
  const float* Hpost = (const float*)d_in[3];  // (4,)
  const float* Hres  = (const float*)d_in[4];  // (4,4)

  const long long total = (long long)in_sizes[0];
  const int B = (int)(total / (long long)(N_EXP * C_DIM));

  float* ws = (float*)d_ws;

  mhc_prep_kernel<<<1, 32, 0, stream>>>(Hpre, Hpost, Hres, ws);
  mhc_main_kernel<<<B, BLOCK_THREADS, 0, stream>>>(x, wgt, ws, (float*)d_out);
}